// DenseNetPWLNN_53171695125391
// MI455X (gfx1250) — compile-verified
//
#include <hip/hip_runtime.h>
#include <hip/hip_bf16.h>

typedef __attribute__((ext_vector_type(2))) float v2f;
typedef __attribute__((ext_vector_type(4))) float v4f;
typedef __attribute__((ext_vector_type(8))) float v8f;

#define N_SMPS    524288
#define WAVES_PB  4
#define BLOCK_T   (WAVES_PB * 32)
#define HSTR      68          // padded LDS row stride (floats) -> conflict-free A-frag b64 loads
#define NB_ROWS   276         // sum of padded K rows over 6 layers: 32+40+44+48+52+60

__device__ __forceinline__ unsigned umin32(unsigned a, unsigned b) { return a < b ? a : b; }
__device__ __forceinline__ unsigned umax32(unsigned a, unsigned b) { return a > b ? a : b; }

// Branchless tanh on v_exp_f32 + v_rcp_f32 (TRANS ops co-execute with the WMMA pipe).
// s = 2^(-2|x|*log2 e) in (0,1] -> no overflow path; saturates for large |x|.
__device__ __forceinline__ float fast_tanh(float v)
{
    float a = __builtin_fabsf(v);
    float s = __builtin_amdgcn_exp2f(a * -2.885390081777927f);   // exp(-2|x|)
    float r = (1.0f - s) * __builtin_amdgcn_rcpf(1.0f + s);
    return __builtin_copysignf(r, v);
}

__global__ __launch_bounds__(BLOCK_T)
void densenet_pwlnn_kernel(const float* __restrict__ x,
                           const float* __restrict__ W0, const float* __restrict__ W1,
                           const float* __restrict__ W2, const float* __restrict__ W3,
                           const float* __restrict__ W4,
                           const float* __restrict__ Wout, const float* __restrict__ bout,
                           const float* __restrict__ ctrs, const float* __restrict__ wts,
                           const float* __restrict__ offs,
                           float* __restrict__ out)
{
    __shared__ float sB[NB_ROWS * 16];            // padded/transposed B tiles, all 6 layers
    __shared__ float sCtr[500];
    __shared__ float sWts[1000];
    __shared__ float sOff[500];
    __shared__ float sBout[2];
    __shared__ float sH[WAVES_PB][32][HSTR];      // per-wave activation tile (+enc in cols 60,61)

    const int tid = threadIdx.x;

    const float* Wl[6] = {W0, W1, W2, W3, W4, Wout};
    const int fanin[6] = {32, 37, 42, 47, 52, 57};
    const int kpad[6]  = {32, 40, 44, 48, 52, 60};
    const int btoff[6] = {0, 32, 72, 116, 164, 216};
    const int noutL[6] = {5, 5, 5, 5, 5, 2};

    // ---- stage weights into zero-padded B tiles: sB[k][n] = (n<nout && k<fanin) ? W[n][k] : 0
    #pragma unroll
    for (int l = 0; l < 6; ++l) {
        const float* W = Wl[l];
        const int f = fanin[l], kp = kpad[l], no = noutL[l];
        float* B = sB + btoff[l] * 16;
        for (int e = tid; e < kp * 16; e += BLOCK_T) {
            int k = e >> 4, n = e & 15;
            B[e] = (n < no && k < f) ? W[n * f + k] : 0.0f;
        }
    }
    for (int e = tid; e < 500;  e += BLOCK_T) sCtr[e] = ctrs[e];
    for (int e = tid; e < 1000; e += BLOCK_T) sWts[e] = wts[e];
    for (int e = tid; e < 500;  e += BLOCK_T) sOff[e] = offs[e];
    if (tid < 2) sBout[tid] = bout[tid];
    __syncthreads();

    const int wave = tid >> 5;
    const int lane = tid & 31;
    const int s0 = (blockIdx.x * WAVES_PB + wave) * 32;
    float (*h)[HSTR] = sH[wave];

    // ---- stage x tile (32 samples x 32 feats), coalesced b128; zero-pad cols 32..59
    {
        const float* xs = x + (size_t)s0 * 32;
        #pragma unroll
        for (int it = 0; it < 8; ++it) {
            int idx = it * 128 + lane * 4;
            v4f v = *(const v4f*)(xs + idx);
            *(v4f*)&h[idx >> 5][idx & 31] = v;
        }
        #pragma unroll
        for (int c = 32; c < 60; c += 4)
            *(v4f*)&h[lane][c] = (v4f){0.f, 0.f, 0.f, 0.f};
    }

    // A-frag mapping (32-bit A 16x4): lane<16 -> K=v, lane>=16 -> K=2+v (ISA 7.12.2)
    const int kk    = (lane >> 4) * 2;
    const int mn    = lane & 15;          // M for A, N for B/D
    const int base8 = (lane >> 4) * 8;    // D rows: VGPR v -> M = v (+8 for hi lanes)

    // ---- dense chain + output projection via fp32 WMMA, K chunked by 4
    #pragma unroll
    for (int l = 0; l < 6; ++l) {
        const int f = fanin[l], kp = kpad[l];
        const float* B = sB + btoff[l] * 16;
        #pragma unroll
        for (int tile = 0; tile < 2; ++tile) {
            const int rb = tile * 16;
            v8f acc = {0.f, 0.f, 0.f, 0.f, 0.f, 0.f, 0.f, 0.f};
            for (int k0 = 0; k0 < kp; k0 += 4) {
                v2f a = *(const v2f*)&h[rb + mn][k0 + kk];
                v2f b;
                b.x = B[(k0 + kk)     * 16 + mn];
                b.y = B[(k0 + kk + 1) * 16 + mn];
                acc = __builtin_amdgcn_wmma_f32_16x16x4_f32(
                          false, a, false, b, (short)0, acc, false, false);
            }
            if (l < 5) {
                if (mn < 5) {                       // only 5 valid output columns
                    #pragma unroll
                    for (int v = 0; v < 8; ++v)
                        h[rb + base8 + v][f + mn] = fast_tanh(acc[v]);
                }
            } else {
                if (mn < 2) {                       // enc = h @ Wout^T + bout -> cols 60,61
                    float bb = sBout[mn];
                    #pragma unroll
                    for (int v = 0; v < 8; ++v)
                        h[rb + base8 + v][60 + mn] = acc[v] + bb;
                }
            }
        }
    }

    // ---- per-lane kNN top-5 over 250 centers (branchless u32 key network)
    float e0 = h[lane][60], e1 = h[lane][61];
    unsigned b0 = 0xFFFFFFFFu, b1 = 0xFFFFFFFFu, b2 = 0xFFFFFFFFu,
             b3 = 0xFFFFFFFFu, b4 = 0xFFFFFFFFu;
    #pragma unroll 5
    for (int j = 0; j < 250; ++j) {
        float c0 = sCtr[2 * j], c1 = sCtr[2 * j + 1];   // LDS broadcast (ds_load_b64)
        float dx = e0 - c0, dy = e1 - c1;
        float d2 = fmaf(dy, dy, dx * dx);               // >= 0 -> bits monotonic as u32
        unsigned key = (__float_as_uint(d2) & 0xFFFFFF00u) | (unsigned)j;
        unsigned t;
        t = umin32(b0, key); key = umax32(b0, key); b0 = t;
        t = umin32(b1, key); key = umax32(b1, key); b1 = t;
        t = umin32(b2, key); key = umax32(b2, key); b2 = t;
        t = umin32(b3, key); key = umax32(b3, key); b3 = t;
        b4 = umin32(b4, key);
    }

    // ---- PWLNN head: sum selected wts/offsets, subtract wts.ctrs, apply to enc
    float w00 = 0.f, w01 = 0.f, w10 = 0.f, w11 = 0.f, o0 = 0.f, o1 = 0.f;
    unsigned sel[5] = {b0, b1, b2, b3, b4};
    #pragma unroll
    for (int t = 0; t < 5; ++t) {
        int j = (int)(sel[t] & 0xFFu);                 // 250 < 256: 8 index bits
        float Wa = sWts[4 * j + 0], Wb = sWts[4 * j + 1];  // wts[j][i][o], flat j*4+i*2+o
        float Wc = sWts[4 * j + 2], Wd = sWts[4 * j + 3];
        float C0 = sCtr[2 * j],     C1 = sCtr[2 * j + 1];
        w00 += Wa; w01 += Wb; w10 += Wc; w11 += Wd;
        o0 += sOff[2 * j]     - (Wa * C0 + Wc * C1);
        o1 += sOff[2 * j + 1] - (Wb * C0 + Wd * C1);
    }
    v2f res;
    res.x = w00 * e0 + w10 * e1 + o0;
    res.y = w01 * e0 + w11 * e1 + o1;
    *(v2f*)(out + (size_t)(s0 + lane) * 2) = res;
}

extern "C" void kernel_launch(void* const* d_in, const int* in_sizes, int n_in,
                              void* d_out, int out_size, void* d_ws, size_t ws_size,
                              hipStream_t stream)
{
    const float* x    = (const float*)d_in[0];
    const float* W0   = (const float*)d_in[1];
    const float* W1   = (const float*)d_in[2];
    const float* W2   = (const float*)d_in[3];
    const float* W3   = (const float*)d_in[4];
    const float* W4   = (const float*)d_in[5];
    const float* Wout = (const float*)d_in[6];
    const float* bout = (const float*)d_in[7];
    const float* ctrs = (const float*)d_in[8];
    const float* wts  = (const float*)d_in[9];
    const float* offs = (const float*)d_in[10];
    float* out = (float*)d_out;

    const int grid = N_SMPS / (WAVES_PB * 32);   // 4096 blocks x 128 threads
    densenet_pwlnn_kernel<<<grid, BLOCK_T, 0, stream>>>(
        x, W0, W1, W2, W3, W4, Wout, bout, ctrs, wts, offs, out);
}